// SAGE_8246337208554
// MI455X (gfx1250) — compile-verified
//
#include <hip/hip_runtime.h>
#include <hip/hip_bf16.h>

#define NODES 50000
#define EDGES 800000
#define DIM   128

typedef __attribute__((ext_vector_type(2))) float v2f;
typedef __attribute__((ext_vector_type(8))) float v8f;

// Native FP32 atomic add (GLOBAL_ATOMIC_ADD_F32) — avoid CAS-loop lowering.
__device__ __forceinline__ void atomAddF32(float* p, float v) {
#if defined(__HIP_DEVICE_COMPILE__)
    unsafeAtomicAdd(p, v);
#else
    atomicAdd(p, v);
#endif
}

// ---------------------------------------------------------------- zero
__global__ void sage_zero_f32(float* __restrict__ p, int n) {
    int i = blockIdx.x * blockDim.x + threadIdx.x;
    if (i < n) p[i] = 0.0f;
}

// ---------------------------------------------------------------- degree (once)
__global__ void sage_degree(const int* __restrict__ row, float* __restrict__ deg, int E) {
    int e = blockIdx.x * blockDim.x + threadIdx.x;
    if (e < E) atomAddF32(&deg[row[e]], 1.0f);
}

// ---------------------------------------------------------------- edge scatter: s[row] += x[col]
// one wave (32 lanes) per edge; each lane handles 4 features
__global__ void sage_scatter(const float* __restrict__ x, const int* __restrict__ row,
                             const int* __restrict__ col, float* __restrict__ s, int E) {
    int t = blockIdx.x * blockDim.x + threadIdx.x;
    int e = t >> 5;
    if (e >= E) return;
    int lane = t & 31;
    const float* __restrict__ src = x + (size_t)col[e] * DIM;
    float* __restrict__ dst = s + (size_t)row[e] * DIM;
#pragma unroll
    for (int j = 0; j < 4; ++j) {
        int f = lane + j * 32;
        atomAddF32(&dst[f], src[f]);
    }
}

// ---------------------------------------------------------------- agg = s / max(deg,1), in place
__global__ void sage_normalize(float* __restrict__ s, const float* __restrict__ deg, int n) {
    int i = blockIdx.x * blockDim.x + threadIdx.x;
    if (i >= n) return;
    float d = deg[i >> 7];  // i / 128
    s[i] = s[i] / fmaxf(d, 1.0f);
}

// ---------------------------------------------------------------- fused GEMM with f32 WMMA
// out = agg @ Wl + bl + x @ Wr + x @ Ws   (all K = 128, DOUT in {128, 64})
// block = 256 threads (8 waves); blockIdx.x = 16-row tile; wave = 16-col tile
template <int DOUT>
__global__ void sage_gemm(const float* __restrict__ agg, const float* __restrict__ x,
                          const float* __restrict__ Wl, const float* __restrict__ bl,
                          const float* __restrict__ Wr, const float* __restrict__ Ws,
                          float* __restrict__ out) {
    constexpr int K = DIM;
    constexpr int nColTiles = DOUT / 16;
    const int wave = threadIdx.x >> 5;
    if (wave >= nColTiles) return;                 // uniform per wave -> EXEC all-1s below
    const int lane = threadIdx.x & 31;
    const int half = lane >> 4;                    // 0: lanes 0-15, 1: lanes 16-31
    const int lidx = lane & 15;
    const int rowBase = blockIdx.x * 16;
    const int col = wave * 16 + lidx;
    const int krow = half * 2;                     // K offset owned by this lane half

    const float* aggRow = agg + (size_t)(rowBase + lidx) * K + krow;
    const float* xRow   = x   + (size_t)(rowBase + lidx) * K + krow;

    v8f c = {};
#pragma unroll 4
    for (int k = 0; k < K; k += 4) {
        // A fragments: 16x4 f32, lane holds (row = lidx, K = k+krow .. k+krow+1)
        v2f aA = *(const v2f*)(aggRow + k);
        v2f aX = *(const v2f*)(xRow + k);
        // B fragments: 4x16 f32, lane holds (K = k+krow .. +1, col)
        v2f bL, bR, bS;
        bL.x = Wl[(size_t)(k + krow) * DOUT + col];
        bL.y = Wl[(size_t)(k + krow + 1) * DOUT + col];
        bR.x = Wr[(size_t)(k + krow) * DOUT + col];
        bR.y = Wr[(size_t)(k + krow + 1) * DOUT + col];
        bS.x = Ws[(size_t)(k + krow) * DOUT + col];
        bS.y = Ws[(size_t)(k + krow + 1) * DOUT + col];
        c = __builtin_amdgcn_wmma_f32_16x16x4_f32(false, aA, false, bL, (short)0, c, false, false);
        c = __builtin_amdgcn_wmma_f32_16x16x4_f32(false, aX, false, bR, (short)0, c, false, false);
        c = __builtin_amdgcn_wmma_f32_16x16x4_f32(false, aX, false, bS, (short)0, c, false, false);
    }

    const float bias = bl[col];
    const int rb = rowBase + half * 8;             // D: VGPR r -> row r (+8 for high half)
#pragma unroll
    for (int r = 0; r < 8; ++r)
        out[(size_t)(rb + r) * DOUT + col] = c[r] + bias;
}

// ---------------------------------------------------------------- per-column sum / sumsq
// block = 128 threads, one column each; each block covers rowsPerBlock rows
__global__ void sage_colstats(const float* __restrict__ pre, float* __restrict__ colsum,
                              float* __restrict__ colsq, int N, int rowsPerBlock) {
    const int c = threadIdx.x;
    int r0 = blockIdx.x * rowsPerBlock;
    int r1 = r0 + rowsPerBlock; if (r1 > N) r1 = N;
    float s = 0.0f, sq = 0.0f;
    for (int r = r0; r < r1; ++r) {
        float v = pre[(size_t)r * DIM + c];
        s += v; sq += v * v;
    }
    atomAddF32(&colsum[c], s);
    atomAddF32(&colsq[c], sq);
}

// ---------------------------------------------------------------- BN + residual + ReLU
__global__ void sage_bn_res_relu(const float* __restrict__ pre, const float* __restrict__ xin,
                                 const float* __restrict__ colsum, const float* __restrict__ colsq,
                                 const float* __restrict__ gamma, const float* __restrict__ beta,
                                 float* __restrict__ xout, int n) {
    int i = blockIdx.x * blockDim.x + threadIdx.x;
    if (i >= n) return;
    int c = i & (DIM - 1);
    const float invN = 1.0f / (float)NODES;
    float mu  = colsum[c] * invN;
    float var = colsq[c] * invN - mu * mu;
    float rs  = rsqrtf(var + 1e-5f);
    float h = (pre[i] - mu) * rs * gamma[c] + beta[c] + xin[i];  // BN + residual (dims match)
    xout[i] = fmaxf(h, 0.0f);
}

// ================================================================ host
extern "C" void kernel_launch(void* const* d_in, const int* in_sizes, int n_in,
                              void* d_out, int out_size, void* d_ws, size_t ws_size,
                              hipStream_t stream) {
    const float* x0  = (const float*)d_in[0];
    const int*   row = (const int*)d_in[1];
    const int*   col = (const int*)d_in[2];
    const float* Wl[3] = {(const float*)d_in[3],  (const float*)d_in[7],  (const float*)d_in[11]};
    const float* bl[3] = {(const float*)d_in[4],  (const float*)d_in[8],  (const float*)d_in[12]};
    const float* Wr[3] = {(const float*)d_in[5],  (const float*)d_in[9],  (const float*)d_in[13]};
    const float* Ws[3] = {(const float*)d_in[6],  (const float*)d_in[10], (const float*)d_in[14]};
    const float* gamma[2] = {(const float*)d_in[15], (const float*)d_in[17]};
    const float* beta[2]  = {(const float*)d_in[16], (const float*)d_in[18]};
    float* out = (float*)d_out;

    const size_t nf = (size_t)NODES * DIM;  // 6.4M floats
    float* ws   = (float*)d_ws;
    float* s    = ws;            // [N,128] aggregation accumulator (normalized in place)
    float* pre  = s + nf;        // [N,128] pre-activations; doubles as layer-1 output x (in-place BN)
    float* xA   = pre + nf;      // [N,128] layer-0 output x
    float* deg  = xA + nf;       // [N]
    float* colsum = deg + NODES; // [128]
    float* colsq  = colsum + DIM;// [128]

    const int B = 256;
    const int gElem   = (int)((nf + B - 1) / B);           // 25000
    const int gEdge   = (EDGES + B - 1) / B;               // 3125
    const int gScat   = (int)(((size_t)EDGES * 32 + B - 1) / B);  // 100000
    const int gGemm   = NODES / 16;                        // 3125 (N divisible by 16)
    const int rowsPB  = 256;
    const int gStats  = (NODES + rowsPB - 1) / rowsPB;     // 196

    // degree: once (row is layer-invariant)
    sage_zero_f32<<<(NODES + B - 1) / B, B, 0, stream>>>(deg, NODES);
    sage_degree<<<gEdge, B, 0, stream>>>(row, deg, EDGES);

    const float* xin[3] = {x0, xA, pre};   // pre aliases layer-1 output
    float* xnext[2] = {xA, pre};

    for (int layer = 0; layer < 2; ++layer) {
        sage_zero_f32<<<gElem, B, 0, stream>>>(s, (int)nf);
        sage_scatter<<<gScat, B, 0, stream>>>(xin[layer], row, col, s, EDGES);
        sage_normalize<<<gElem, B, 0, stream>>>(s, deg, (int)nf);
        sage_gemm<DIM><<<gGemm, B, 0, stream>>>(s, xin[layer], Wl[layer], bl[layer],
                                                Wr[layer], Ws[layer], pre);
        sage_zero_f32<<<1, 256, 0, stream>>>(colsum, 2 * DIM);  // colsum+colsq contiguous
        sage_colstats<<<gStats, DIM, 0, stream>>>(pre, colsum, colsq, NODES, rowsPB);
        sage_bn_res_relu<<<gElem, B, 0, stream>>>(pre, xin[layer], colsum, colsq,
                                                  gamma[layer], beta[layer],
                                                  xnext[layer], (int)nf);
    }

    // final layer: h = agg @ Wl2 + bl2 + x @ Wr2 + x @ Ws2  -> d_out [N, 64]
    sage_zero_f32<<<gElem, B, 0, stream>>>(s, (int)nf);
    sage_scatter<<<gScat, B, 0, stream>>>(xin[2], row, col, s, EDGES);
    sage_normalize<<<gElem, B, 0, stream>>>(s, deg, (int)nf);
    sage_gemm<64><<<gGemm, B, 0, stream>>>(s, xin[2], Wl[2], bl[2], Wr[2], Ws[2], out);
}